// SteinKernel_30150670418088
// MI455X (gfx1250) — compile-verified
//
#include <hip/hip_runtime.h>
#include <math.h>

#define N_ROWS 8192
#define M_ROWS 8192
#define DDIM   64
#define SAMP   512

typedef __attribute__((ext_vector_type(2))) float v2f;
typedef __attribute__((ext_vector_type(8))) float v8f;

// ---------------------------------------------------------------------------
// Workspace layout (floats):
//   [0..15]            params: [0]=1/(2h^2) [1]=1/h^2 [2]=1/h^4
//   [16..16+8191]      x2[i]   = sum_d x[i,d]^2
//   [+8192]            y2[j]   = sum_d y[j,d]^2
//   [+8192]            ysy[j]  = sum_d y[j,d]*sy[j,d]
//   [32768..32768+262143]  dsq sample (512*512)
// total ~1.13 MB
// ---------------------------------------------------------------------------

__global__ void rowsums_kernel(const float* __restrict__ x,
                               const float* __restrict__ y,
                               const float* __restrict__ sy,
                               float* __restrict__ x2,
                               float* __restrict__ y2,
                               float* __restrict__ ysy) {
  int tid = blockIdx.x * blockDim.x + threadIdx.x;
  if (tid < N_ROWS) {
    const float4* p = (const float4*)(x + (size_t)tid * DDIM);
    float s = 0.f;
#pragma unroll
    for (int q = 0; q < DDIM / 4; ++q) {
      float4 v = p[q];
      s += v.x * v.x + v.y * v.y + v.z * v.z + v.w * v.w;
    }
    x2[tid] = s;
  } else {
    int r = tid - N_ROWS;
    const float4* py = (const float4*)(y  + (size_t)r * DDIM);
    const float4* ps = (const float4*)(sy + (size_t)r * DDIM);
    float s2 = 0.f, sd = 0.f;
#pragma unroll
    for (int q = 0; q < DDIM / 4; ++q) {
      float4 a = py[q], b = ps[q];
      s2 += a.x * a.x + a.y * a.y + a.z * a.z + a.w * a.w;
      sd += a.x * b.x + a.y * b.y + a.z * b.z + a.w * b.w;
    }
    y2[r]  = s2;
    ysy[r] = sd;
  }
}

__global__ void sample_dsq_kernel(const float* __restrict__ x,
                                  const float* __restrict__ y,
                                  float* __restrict__ dsq) {
  int idx = blockIdx.x * blockDim.x + threadIdx.x;  // 0..262143
  int i = idx >> 9;
  int j = idx & 511;
  const float4* a = (const float4*)(x + (size_t)i * DDIM);
  const float4* b = (const float4*)(y + (size_t)j * DDIM);
  float s = 0.f;
#pragma unroll
  for (int q = 0; q < DDIM / 4; ++q) {
    float4 va = a[q], vb = b[q];
    float d0 = va.x - vb.x, d1 = va.y - vb.y;
    float d2 = va.z - vb.z, d3 = va.w - vb.w;
    s += d0 * d0 + d1 * d1 + d2 * d2 + d3 * d3;
  }
  dsq[idx] = s;
}

// Exact lower median of SAMP*SAMP non-negative floats via MSB-first radix
// selection on the uint bit pattern (monotone for x >= 0). Single workgroup.
__global__ void median_kernel(const float* __restrict__ dsq,
                              float* __restrict__ params) {
  __shared__ unsigned total;
  const unsigned K = (SAMP * SAMP - 1) / 2;  // lower median, 0-indexed
  unsigned prefix = 0u;
  for (int bit = 30; bit >= 0; --bit) {
    unsigned cand = prefix | (1u << bit);
    if (threadIdx.x == 0) total = 0u;
    __syncthreads();
    unsigned local = 0u;
    for (int idx = threadIdx.x; idx < SAMP * SAMP; idx += blockDim.x) {
      unsigned v = __float_as_uint(dsq[idx]);
      local += (v < cand) ? 1u : 0u;
    }
    atomicAdd(&total, local);
    __syncthreads();
    if (total <= K) prefix = cand;  // k-th smallest >= cand
    __syncthreads();
  }
  if (threadIdx.x == 0) {
    float med = __uint_as_float(prefix);
    float h = sqrtf(med * 0.5f);            // BW_FACTOR == 1.0
    h = fminf(fmaxf(h, 0.1f), 10.0f);       // clip [MIN_BW, MAX_BW]
    float h2 = h * h;
    params[0] = 1.0f / (2.0f * h2);
    params[1] = 1.0f / h2;
    params[2] = 1.0f / (h2 * h2);
  }
}

// ---------------------------------------------------------------------------
// Main fused kernel: 8 waves/block; wave computes a 16(rows) x 64(cols) strip
// = 4 tiles of 16x16, with three fp32 WMMA accumulators per tile:
//   AB = x . y^T,  S = sx . sy^T,  XS = x . sy^T
// A-frag layout (32-bit A 16x4): lane<16 -> row=lane, K = k0,k0+1
//                                lane>=16 -> row=lane-16, K = k0+2,k0+3
// B-frag (4x16): lane<16 -> col=lane, K = k0,k0+1; lane>=16 -> K = k0+2,k0+3
// C/D (16x16 f32): VGPR g: lanes0-15 M=g, lanes16-31 M=g+8; N = lane&15
// ---------------------------------------------------------------------------
__global__ __launch_bounds__(256) void stein_main_kernel(
    const float* __restrict__ x,  const float* __restrict__ sx,
    const float* __restrict__ y,  const float* __restrict__ sy,
    const float* __restrict__ x2, const float* __restrict__ y2,
    const float* __restrict__ ysy, const float* __restrict__ params,
    float* __restrict__ out) {
  const int lane = threadIdx.x & 31;
  const int wave = threadIdx.x >> 5;
  const int rowBase = blockIdx.y * 128 + wave * 16;  // N (x rows)
  const int colBase = blockIdx.x * 64;               // M (y rows)

  const int r    = lane & 15;
  const int half = lane >> 4;

  v8f accAB[4], accS[4], accXS[4];
  const v8f vzero = {0.f, 0.f, 0.f, 0.f, 0.f, 0.f, 0.f, 0.f};
#pragma unroll
  for (int c = 0; c < 4; ++c) {
    accAB[c] = vzero;
    accS[c]  = vzero;
    accXS[c] = vzero;
  }

  const float* xrow  = x  + (size_t)(rowBase + r) * DDIM;
  const float* sxrow = sx + (size_t)(rowBase + r) * DDIM;

#pragma unroll 4
  for (int k0 = 0; k0 < DDIM; k0 += 4) {
    const int koff = k0 + 2 * half;
    v2f ax  = *(const v2f*)(xrow  + koff);
    v2f asx = *(const v2f*)(sxrow + koff);
#pragma unroll
    for (int c = 0; c < 4; ++c) {
      const size_t jrow = (size_t)(colBase + c * 16 + r) * DDIM;
      v2f by  = *(const v2f*)(y  + jrow + koff);
      v2f bsy = *(const v2f*)(sy + jrow + koff);
      accAB[c] = __builtin_amdgcn_wmma_f32_16x16x4_f32(
          false, ax, false, by, (short)0, accAB[c], false, false);
      accS[c] = __builtin_amdgcn_wmma_f32_16x16x4_f32(
          false, asx, false, bsy, (short)0, accS[c], false, false);
      accXS[c] = __builtin_amdgcn_wmma_f32_16x16x4_f32(
          false, ax, false, bsy, (short)0, accXS[c], false, false);
    }
  }

  const float inv2h2 = params[0];
  const float invh2  = params[1];
  const float invh4  = params[2];
  const float dOverH2 = (float)DDIM * invh2;

  float x2v[8];
#pragma unroll
  for (int g = 0; g < 8; ++g) x2v[g] = x2[rowBase + g + 8 * half];

#pragma unroll
  for (int c = 0; c < 4; ++c) {
    const int j = colBase + c * 16 + r;
    const float y2j  = y2[j];
    const float ysyj = ysy[j];
#pragma unroll
    for (int g = 0; g < 8; ++g) {
      const int i = rowBase + g + 8 * half;
      float ab   = accAB[c][g];
      float dist = fmaxf(x2v[g] + y2j - 2.0f * ab, 0.0f);
      float Kf   = __expf(-dist * inv2h2);
      float t1   = Kf * accS[c][g];
      float t2   = (accXS[c][g] - ysyj) * Kf * invh2;
      float t3   = dist * Kf * invh4 - Kf * dOverH2;
      out[(size_t)i * M_ROWS + j] = t1 + t2 + t3;
    }
  }
}

extern "C" void kernel_launch(void* const* d_in, const int* in_sizes, int n_in,
                              void* d_out, int out_size, void* d_ws,
                              size_t ws_size, hipStream_t stream) {
  (void)in_sizes; (void)n_in; (void)out_size; (void)ws_size;
  const float* x  = (const float*)d_in[0];
  const float* sx = (const float*)d_in[1];
  const float* y  = (const float*)d_in[2];
  const float* sy = (const float*)d_in[3];
  float* out = (float*)d_out;

  float* wsf    = (float*)d_ws;
  float* params = wsf;            // 16 floats
  float* x2     = wsf + 16;       // 8192
  float* y2     = x2 + N_ROWS;    // 8192
  float* ysy    = y2 + M_ROWS;    // 8192
  float* dsq    = wsf + 32768;    // 262144

  rowsums_kernel<<<(N_ROWS + M_ROWS) / 256, 256, 0, stream>>>(x, y, sy, x2, y2,
                                                              ysy);
  sample_dsq_kernel<<<(SAMP * SAMP) / 256, 256, 0, stream>>>(x, y, dsq);
  median_kernel<<<1, 1024, 0, stream>>>(dsq, params);

  dim3 grid(M_ROWS / 64, N_ROWS / 128);
  stein_main_kernel<<<grid, 256, 0, stream>>>(x, sx, y, sy, x2, y2, ysy, params,
                                              out);
}